// ScaledDotProductAttention_70291434766698
// MI455X (gfx1250) — compile-verified
//
#include <hip/hip_runtime.h>
#include <hip/hip_bf16.h>

typedef __attribute__((ext_vector_type(2))) float v2f;
typedef __attribute__((ext_vector_type(8))) float v8f;

#define NB   8
#define LSEQ 2048
#define DDIM 768
#define HDIM 512
#define GROWS 8

__device__ __forceinline__ v8f wmma_f32(v2f a, v2f b, v8f c) {
  // (neg_a, A, neg_b, B, c_mod, C, reuse_a, reuse_b)
  return __builtin_amdgcn_wmma_f32_16x16x4_f32(false, a, false, b, (short)0, c,
                                               false, false);
}

// ---------------------------------------------------------------------------
// Kernel 1: gate g[b,l] = tanh(v[b,l,:] @ W1 + b1) @ W2 + b2
// 8 v-rows per block so W1 (1.5MB, L2-resident) is amortized 8x.
// ---------------------------------------------------------------------------
__global__ __launch_bounds__(512)
void gate_kernel(const float* __restrict__ v, const float* __restrict__ W1,
                 const float* __restrict__ b1, const float* __restrict__ W2,
                 const float* __restrict__ b2, float* __restrict__ g) {
  __shared__ float vs[GROWS * DDIM];
  __shared__ float red[512];
  const int tid = threadIdx.x;
  const long grp = blockIdx.x;
  const float* vbase = v + grp * (long)(GROWS * DDIM);
  for (int i = tid; i < GROWS * DDIM; i += 512) vs[i] = vbase[i];
  __syncthreads();

  float acc[GROWS];
  const float bias = b1[tid];
#pragma unroll
  for (int r = 0; r < GROWS; ++r) acc[r] = bias;
  for (int d = 0; d < DDIM; ++d) {
    const float w = W1[d * HDIM + tid];
#pragma unroll
    for (int r = 0; r < GROWS; ++r) acc[r] += vs[r * DDIM + d] * w;
  }
  const float w2  = W2[tid];
  const float b2v = b2[0];
#pragma unroll
  for (int r = 0; r < GROWS; ++r) acc[r] = tanhf(acc[r]) * w2;

  for (int r = 0; r < GROWS; ++r) {
    red[tid] = acc[r];
    __syncthreads();
    for (int s = 256; s > 0; s >>= 1) {
      if (tid < s) red[tid] += red[tid + s];
      __syncthreads();
    }
    if (tid == 0) g[grp * GROWS + r] = red[0] + b2v;
    __syncthreads();
  }
}

// ---------------------------------------------------------------------------
// Kernel 2: S[b,q,k] = dot(q_row, k_row) / sqrt(L).
// 128x64 tile per WG; 8 waves, each owns a 2x2 block of 16x16 WMMA subtiles
// (4 x v8f accumulators) -> 4 WMMAs per 4 LDS fragment loads.
// ---------------------------------------------------------------------------
__global__ __launch_bounds__(256)
void scores_kernel(const float* __restrict__ q, const float* __restrict__ k,
                   float* __restrict__ S) {
  __shared__ float Qs[128][18];  // stride 18 floats = 72B (8B aligned)
  __shared__ float Ks[64][18];
  const int tid = threadIdx.x;
  const int wave = tid >> 5, lane = tid & 31;
  const int lane15 = lane & 15;
  const int kb = (lane >> 4) << 1;
  const int bm = blockIdx.x * 128, bn = blockIdx.y * 64;
  const long b = blockIdx.z;
  const float* Qb = q + b * (long)LSEQ * DDIM;
  const float* Kb = k + b * (long)LSEQ * DDIM;
  float* Sb = S + b * (long)LSEQ * LSEQ;

  const int qldr = tid >> 1;        // 0..127
  const int qldc = (tid & 1) << 3;  // 0 or 8
  const int kldr = tid >> 2;        // 0..63
  const int kldc = (tid & 3) << 2;  // 0,4,8,12

  const int mA0 = (wave & 3) * 32 + lane15;   // two M subtiles per wave
  const int mA1 = mA0 + 16;
  const int nB0 = (wave >> 2) * 32 + lane15;  // two N subtiles per wave
  const int nB1 = nB0 + 16;

  v8f acc00 = {}, acc01 = {}, acc10 = {}, acc11 = {};
  for (int kc = 0; kc < DDIM; kc += 16) {
    const float* qsrc = Qb + (long)(bm + qldr) * DDIM + kc + qldc;
    const float4 qv0 = *(const float4*)(qsrc);
    const float4 qv1 = *(const float4*)(qsrc + 4);
    const float4 kv = *(const float4*)(Kb + (long)(bn + kldr) * DDIM + kc + kldc);
    Qs[qldr][qldc+0] = qv0.x; Qs[qldr][qldc+1] = qv0.y;
    Qs[qldr][qldc+2] = qv0.z; Qs[qldr][qldc+3] = qv0.w;
    Qs[qldr][qldc+4] = qv1.x; Qs[qldr][qldc+5] = qv1.y;
    Qs[qldr][qldc+6] = qv1.z; Qs[qldr][qldc+7] = qv1.w;
    Ks[kldr][kldc+0] = kv.x; Ks[kldr][kldc+1] = kv.y;
    Ks[kldr][kldc+2] = kv.z; Ks[kldr][kldc+3] = kv.w;
    __syncthreads();
#pragma unroll
    for (int kk = 0; kk < 16; kk += 4) {
      v2f a0, a1, b0, b1;
      a0.x = Qs[mA0][kk + kb]; a0.y = Qs[mA0][kk + kb + 1];
      a1.x = Qs[mA1][kk + kb]; a1.y = Qs[mA1][kk + kb + 1];
      b0.x = Ks[nB0][kk + kb]; b0.y = Ks[nB0][kk + kb + 1];
      b1.x = Ks[nB1][kk + kb]; b1.y = Ks[nB1][kk + kb + 1];
      acc00 = wmma_f32(a0, b0, acc00);
      acc01 = wmma_f32(a0, b1, acc01);
      acc10 = wmma_f32(a1, b0, acc10);
      acc11 = wmma_f32(a1, b1, acc11);
    }
    __syncthreads();
  }
  const float scale = 0.022097086912079608f;  // 1/sqrt(2048)
  const int r0 = bm + (wave & 3) * 32 + ((lane >> 4) << 3);
  const int c0 = bn + (wave >> 2) * 32 + lane15;
#pragma unroll
  for (int i = 0; i < 8; ++i) {
    Sb[(long)(r0 + i) * LSEQ + c0]           = acc00[i] * scale;
    Sb[(long)(r0 + i) * LSEQ + c0 + 16]      = acc01[i] * scale;
    Sb[(long)(r0 + 16 + i) * LSEQ + c0]      = acc10[i] * scale;
    Sb[(long)(r0 + 16 + i) * LSEQ + c0 + 16] = acc11[i] * scale;
  }
}

// ---------------------------------------------------------------------------
// Kernel 3: per-(b,k) column-wise online max / sum-exp (softmax over axis=1)
// ---------------------------------------------------------------------------
__global__ __launch_bounds__(256)
void colstats_kernel(const float* __restrict__ S, float* __restrict__ M,
                     float* __restrict__ Z) {
  const long b = blockIdx.y;
  const int kidx = blockIdx.x * 256 + threadIdx.x;
  const float* col = S + b * (long)LSEQ * LSEQ + kidx;
  float m = -3.402823466e38f, z = 0.f;
  for (int qi = 0; qi < LSEQ; ++qi) {
    const float x  = col[(long)qi * LSEQ];
    const float mn = fmaxf(m, x);
    z = z * expf(m - mn) + expf(x - mn);
    m = mn;
  }
  M[b * LSEQ + kidx] = m;
  Z[b * LSEQ + kidx] = z;
}

// ---------------------------------------------------------------------------
// Kernel 4: per-row: s = exp(S - M)/Z; a = (s*(1-g) + g/L)/sqrt(768);
//           attn_weights = softmax_k(a), in place.
// ---------------------------------------------------------------------------
__global__ __launch_bounds__(256)
void rowsoftmax_kernel(float* __restrict__ S, const float* __restrict__ g,
                       const float* __restrict__ M, const float* __restrict__ Z) {
  __shared__ float red[256];
  const int tid = threadIdx.x;
  const long b = blockIdx.y;
  const long qrow = blockIdx.x;
  float* row = S + (b * LSEQ + qrow) * (long)LSEQ;
  const float* gp = g + b * LSEQ;
  const float* Mp = M + b * LSEQ;
  const float* Zp = Z + b * LSEQ;
  const float cmean    = 1.0f / (float)LSEQ;    // mean over axis-1 softmax == 1/L
  const float invScale = 0.03608439182435161f;  // 1/sqrt(768)

  float e[8];
  float rmax = -3.402823466e38f;
#pragma unroll
  for (int j = 0; j < 8; ++j) {
    const int kk = tid + j * 256;
    const float sv = expf(row[kk] - Mp[kk]) / Zp[kk];
    const float gv = gp[kk];
    const float a  = (sv * (1.f - gv) + cmean * gv) * invScale;
    e[j] = a;
    rmax = fmaxf(rmax, a);
  }
  red[tid] = rmax; __syncthreads();
  for (int s = 128; s > 0; s >>= 1) {
    if (tid < s) red[tid] = fmaxf(red[tid], red[tid + s]);
    __syncthreads();
  }
  const float rm = red[0]; __syncthreads();
  float rsum = 0.f;
#pragma unroll
  for (int j = 0; j < 8; ++j) { e[j] = expf(e[j] - rm); rsum += e[j]; }
  red[tid] = rsum; __syncthreads();
  for (int s = 128; s > 0; s >>= 1) {
    if (tid < s) red[tid] += red[tid + s];
    __syncthreads();
  }
  const float inv = 1.f / red[0];
#pragma unroll
  for (int j = 0; j < 8; ++j) row[tid + j * 256] = e[j] * inv;
}

// ---------------------------------------------------------------------------
// Kernel 5: attn_output = attn_weights @ V + Q
// 128x64 tile per WG, 2x2 WMMA subtiles per wave.
// ---------------------------------------------------------------------------
__global__ __launch_bounds__(256)
void outgemm_kernel(const float* __restrict__ W, const float* __restrict__ v,
                    const float* __restrict__ q, float* __restrict__ out) {
  __shared__ float Ws[128][18];
  __shared__ float Vs[16][68];
  const int tid = threadIdx.x;
  const int wave = tid >> 5, lane = tid & 31;
  const int lane15 = lane & 15;
  const int kb = (lane >> 4) << 1;
  const int bm = blockIdx.x * 128;  // query rows
  const int bn = blockIdx.y * 64;   // embed cols
  const long b = blockIdx.z;
  const float* Wb = W + b * (long)LSEQ * LSEQ;
  const float* Vb = v + b * (long)LSEQ * DDIM;
  const float* Qb = q + b * (long)LSEQ * DDIM;
  float* Ob = out + b * (long)LSEQ * DDIM;

  const int wldr = tid >> 1;        // 0..127
  const int wldc = (tid & 1) << 3;  // 0 or 8
  const int vldr = tid >> 4;        // 0..15
  const int vldc = (tid & 15) << 2; // 0..60

  const int mA0 = (wave & 3) * 32 + lane15;
  const int mA1 = mA0 + 16;
  const int nB0 = (wave >> 2) * 32 + lane15;  // 0..63 within V tile
  const int nB1 = nB0 + 16;

  v8f acc00 = {}, acc01 = {}, acc10 = {}, acc11 = {};
  for (int kc = 0; kc < LSEQ; kc += 16) {
    const float* wsrc = Wb + (long)(bm + wldr) * LSEQ + kc + wldc;
    const float4 wv0 = *(const float4*)(wsrc);
    const float4 wv1 = *(const float4*)(wsrc + 4);
    const float4 vv = *(const float4*)(Vb + (long)(kc + vldr) * DDIM + bn + vldc);
    Ws[wldr][wldc+0] = wv0.x; Ws[wldr][wldc+1] = wv0.y;
    Ws[wldr][wldc+2] = wv0.z; Ws[wldr][wldc+3] = wv0.w;
    Ws[wldr][wldc+4] = wv1.x; Ws[wldr][wldc+5] = wv1.y;
    Ws[wldr][wldc+6] = wv1.z; Ws[wldr][wldc+7] = wv1.w;
    Vs[vldr][vldc+0] = vv.x; Vs[vldr][vldc+1] = vv.y;
    Vs[vldr][vldc+2] = vv.z; Vs[vldr][vldc+3] = vv.w;
    __syncthreads();
#pragma unroll
    for (int kk = 0; kk < 16; kk += 4) {
      v2f a0, a1, b0, b1;
      a0.x = Ws[mA0][kk + kb];  a0.y = Ws[mA0][kk + kb + 1];
      a1.x = Ws[mA1][kk + kb];  a1.y = Ws[mA1][kk + kb + 1];
      b0.x = Vs[kk + kb][nB0];  b0.y = Vs[kk + kb + 1][nB0];
      b1.x = Vs[kk + kb][nB1];  b1.y = Vs[kk + kb + 1][nB1];
      acc00 = wmma_f32(a0, b0, acc00);
      acc01 = wmma_f32(a0, b1, acc01);
      acc10 = wmma_f32(a1, b0, acc10);
      acc11 = wmma_f32(a1, b1, acc11);
    }
    __syncthreads();
  }
  const int r0 = bm + (wave & 3) * 32 + ((lane >> 4) << 3);
  const int c0 = bn + (wave >> 2) * 32 + lane15;
#pragma unroll
  for (int i = 0; i < 8; ++i) {
    const long ra = r0 + i;
    const long rb = r0 + 16 + i;
    Ob[ra * DDIM + c0]      = acc00[i] + Qb[ra * DDIM + c0];
    Ob[ra * DDIM + c0 + 16] = acc01[i] + Qb[ra * DDIM + c0 + 16];
    Ob[rb * DDIM + c0]      = acc10[i] + Qb[rb * DDIM + c0];
    Ob[rb * DDIM + c0 + 16] = acc11[i] + Qb[rb * DDIM + c0 + 16];
  }
}

// ---------------------------------------------------------------------------
extern "C" void kernel_launch(void* const* d_in, const int* in_sizes, int n_in,
                              void* d_out, int out_size, void* d_ws, size_t ws_size,
                              hipStream_t stream) {
  const float* q  = (const float*)d_in[0];
  const float* k  = (const float*)d_in[1];
  const float* v  = (const float*)d_in[2];
  const float* W1 = (const float*)d_in[3];
  const float* b1 = (const float*)d_in[4];
  const float* W2 = (const float*)d_in[5];
  const float* b2 = (const float*)d_in[6];

  float* out      = (float*)d_out;
  float* attn_out = out;                            // NB*LSEQ*DDIM floats
  float* Wbuf     = out + (long)NB * LSEQ * DDIM;   // NB*LSEQ*LSEQ floats (weights out)

  // Small per-key stats (g, M, Z): 3 * NB * LSEQ floats.
  const size_t need = (size_t)3 * NB * LSEQ * sizeof(float);
  float* scratch = (ws_size >= need) ? (float*)d_ws
                                     : attn_out;    // attn_out written only by last kernel
  float* gbuf = scratch;
  float* Mbuf = scratch + (long)NB * LSEQ;
  float* Zbuf = scratch + (long)2 * NB * LSEQ;

  gate_kernel<<<dim3(NB * LSEQ / GROWS), 512, 0, stream>>>(v, W1, b1, W2, b2, gbuf);
  scores_kernel<<<dim3(LSEQ / 128, LSEQ / 64, NB), 256, 0, stream>>>(q, k, Wbuf);
  colstats_kernel<<<dim3(LSEQ / 256, NB), 256, 0, stream>>>(Wbuf, Mbuf, Zbuf);
  rowsoftmax_kernel<<<dim3(LSEQ, NB), 256, 0, stream>>>(Wbuf, gbuf, Mbuf, Zbuf);
  outgemm_kernel<<<dim3(LSEQ / 128, DDIM / 64, NB), 256, 0, stream>>>(Wbuf, v, q, attn_out);
}